// Cheb_23880018166265
// MI455X (gfx1250) — compile-verified
//
#include <hip/hip_runtime.h>
#include <hip/hip_bf16.h>
#include <math.h>

typedef float v2f __attribute__((ext_vector_type(2)));
typedef float v8f __attribute__((ext_vector_type(8)));

#define THREADS 256
static const int F_IN = 128, HID = 128, HID4 = 32, OUTF = 40, OUTP = 48, KCH = 5;

// ---------------- elementwise / init kernels ----------------

__global__ void zero_f32(float* __restrict__ p, size_t n) {
  size_t i = (size_t)blockIdx.x * blockDim.x + threadIdx.x;
  if (i < n) p[i] = 0.0f;
}

__global__ void zero_f64(double* __restrict__ p, int n) {
  int i = blockIdx.x * blockDim.x + threadIdx.x;
  if (i < n) p[i] = 0.0;
}

__global__ void negcopy_f32(float* __restrict__ o, const float* __restrict__ s, size_t n) {
  size_t i = (size_t)blockIdx.x * blockDim.x + threadIdx.x;
  if (i < n) o[i] = -s[i];
}

// ---------------- feature standardization (coalesced, f64 atomics) ----------------
// blockDim.x == F (128): thread = column, blocks partition rows -> coalesced reads.

__global__ void stats_partial(const float* __restrict__ x, double* __restrict__ gs,
                              double* __restrict__ gq, int N, int F) {
  int c = threadIdx.x;
  double s = 0.0, q = 0.0;
  for (int r = blockIdx.x; r < N; r += gridDim.x) {
    float v = x[(size_t)r * F + c];
    s += (double)v;
    q += (double)v * (double)v;
  }
  atomicAdd(&gs[c], s);   // global_atomic_add_f64
  atomicAdd(&gq[c], q);
}

__global__ void stats_final(const double* __restrict__ gs, const double* __restrict__ gq,
                            float* __restrict__ mean, float* __restrict__ rstd, int N) {
  int c = threadIdx.x;
  double sum = gs[c], sumsq = gq[c];
  double m = sum / (double)N;
  double var = (sumsq - sum * sum / (double)N) / (double)(N - 1);  // ddof=1
  mean[c] = (float)m;
  rstd[c] = (float)(1.0 / sqrt(var));
}

__global__ void standardize(const float* __restrict__ x, const float* __restrict__ mean,
                            const float* __restrict__ rstd, float* __restrict__ o,
                            size_t n, int Fmask) {
  size_t i = (size_t)blockIdx.x * blockDim.x + threadIdx.x;
  if (i < n) {
    int c = (int)i & Fmask;
    o[i] = (x[i] - mean[c]) * rstd[c];
  }
}

// ---------------- graph normalization ----------------

__global__ void degree_k(const int* __restrict__ src, float* __restrict__ deg, int E) {
  int e = blockIdx.x * blockDim.x + threadIdx.x;
  if (e < E) atomicAdd(&deg[src[e]], 1.0f);
}

__global__ void dis_k(float* __restrict__ deg, int N) {  // in-place deg -> dis
  int i = blockIdx.x * blockDim.x + threadIdx.x;
  if (i < N) {
    float d = deg[i];
    deg[i] = (d > 0.0f) ? rsqrtf(fmaxf(d, 1.0f)) : 0.0f;
  }
}

__global__ void wedge_k(const int* __restrict__ src, const int* __restrict__ dst,
                        const float* __restrict__ dis, float* __restrict__ w, int E) {
  int e = blockIdx.x * blockDim.x + threadIdx.x;
  if (e < E) w[e] = -dis[src[e]] * dis[dst[e]];
}

// ---------------- W3 zero-padding: [K][32][40] -> [K][32][48] ----------------

__global__ void pad_w3(const float* __restrict__ w, float* __restrict__ wp, int nRows) {
  int i = blockIdx.x * blockDim.x + threadIdx.x;   // over nRows*OUTP
  if (i < nRows * OUTP) {
    int row = i / OUTP, col = i - row * OUTP;
    wp[i] = (col < OUTF) ? w[row * OUTF + col] : 0.0f;
  }
}

// ---------------- sparse propagation: o[dst] += scale*w*t[src] ----------------
// one wave32 per edge; lane handles 4 contiguous features (float4 gather)

__global__ void scatter_k(const float* __restrict__ t, float* __restrict__ o,
                          const int* __restrict__ src, const int* __restrict__ dst,
                          const float* __restrict__ w, int F, float scale, int E) {
  int wid = (int)(((size_t)blockIdx.x * blockDim.x + threadIdx.x) >> 5);
  int lane = threadIdx.x & 31;
  if (wid >= E) return;                      // wave-uniform
  int s = src[wid], d = dst[wid];
  float we = w[wid] * scale;
  int nch = F >> 2;
  const float4* tv = (const float4*)(t + (size_t)s * F);
  float* ob = o + (size_t)d * F;
  for (int c = lane; c < nch; c += 32) {
    float4 v = tv[c];
    float* p = ob + (size_t)c * 4;
    atomicAdd(p + 0, we * v.x);
    atomicAdd(p + 1, we * v.y);
    atomicAdd(p + 2, we * v.z);
    atomicAdd(p + 3, we * v.w);
  }
}

// ---------------- fp32 WMMA GEMM: C[N x Fout] (+)= A[N x Fin] * W[Fin x Fout] ----
// One wave per 16x16 output tile; V_WMMA_F32_16X16X4_F32 (W pre-padded: no predication).
// A layout (16x4 f32): lanes 0-15 -> M=lane, K={k0,k0+1}; lanes 16-31 -> M=lane-16, K={k0+2,k0+3}
// B layout (4x16 f32): VGPR j, lanes 0-15 -> K=j, N=lane; lanes 16-31 -> K=j+2, N=lane-16
// C/D layout: VGPR i, lanes 0-15 -> M=i, lanes 16-31 -> M=i+8, N=lane%16

__global__ void gemm_wmma_f32(const float* __restrict__ A, const float* __restrict__ W,
                              float* __restrict__ C, int N, int Fin, int Fout,
                              int accumulate) {
  int wave = (int)(((size_t)blockIdx.x * blockDim.x + threadIdx.x) >> 5);
  int tilesN = Fout >> 4;
  int totalTiles = (N >> 4) * tilesN;
  if (wave >= totalTiles) return;            // wave-uniform: EXEC stays all-1s
  int tm = wave / tilesN, tn = wave - tm * tilesN;
  int row0 = tm << 4, col0 = tn << 4;
  int lane = threadIdx.x & 31;
  int half = lane >> 4;                      // 0 or 1
  int l = lane & 15;
  int colIdx = col0 + l;

  v8f c = {};
  if (accumulate) {
#pragma unroll
    for (int i = 0; i < 8; ++i)
      c[i] = C[(size_t)(row0 + i + 8 * half) * Fout + colIdx];
  }

  const float* ap = A + (size_t)(row0 + l) * Fin + (half << 1);
  const float* wp = W + (size_t)(half << 1) * Fout + colIdx;
  const size_t wstep = (size_t)Fout * 4;

#pragma unroll 4
  for (int k0 = 0; k0 < Fin; k0 += 4) {
    v2f a = *(const v2f*)ap;                 // contiguous pair: 8B load
    v2f b;
    b.x = wp[0];
    b.y = wp[Fout];
    c = __builtin_amdgcn_wmma_f32_16x16x4_f32(false, a, false, b, (short)0, c,
                                              false, false);
    ap += 4;
    wp += wstep;
  }

#pragma unroll
  for (int i = 0; i < 8; ++i)
    C[(size_t)(row0 + i + 8 * half) * Fout + colIdx] = c[i];
}

// ---------------- epilogues ----------------

__global__ void bias_relu(float* __restrict__ h, const float* __restrict__ b,
                          size_t n, int Fmask) {
  size_t i = (size_t)blockIdx.x * blockDim.x + threadIdx.x;
  if (i < n) {
    int c = (int)i & Fmask;
    h[i] = fmaxf(h[i] + b[c], 0.0f);
  }
}

__global__ void final_out(const float* __restrict__ hp, const float* __restrict__ b,
                          float* __restrict__ out, int N) {
  size_t i = (size_t)blockIdx.x * blockDim.x + threadIdx.x;
  size_t n = (size_t)N * OUTF;
  if (i < n) {
    int row = (int)(i / OUTF);
    int col = (int)(i - (size_t)row * OUTF);
    out[i] = hp[(size_t)row * OUTP + col] + b[col];
  }
}

// ---------------- host orchestration ----------------

static inline unsigned gblk(size_t n) { return (unsigned)((n + THREADS - 1) / THREADS); }

extern "C" void kernel_launch(void* const* d_in, const int* in_sizes, int n_in,
                              void* d_out, int out_size, void* d_ws, size_t ws_size,
                              hipStream_t stream) {
  const float* x   = (const float*)d_in[0];
  const int*   ei  = (const int*)d_in[1];
  const float* W1  = (const float*)d_in[2];
  const float* b1  = (const float*)d_in[3];
  const float* W2  = (const float*)d_in[4];
  const float* b2  = (const float*)d_in[5];
  const float* W3  = (const float*)d_in[6];
  const float* b3  = (const float*)d_in[7];
  float* out = (float*)d_out;

  const int N = in_sizes[0] / F_IN;
  const int E = in_sizes[1] / 2;
  const int* src = ei;
  const int* dst = ei + E;

  // workspace layout (floats; all region starts stay 8B-aligned)
  const size_t NB = (size_t)N * F_IN;
  float* bufA   = (float*)d_ws;           // xn, then recurrence temp
  float* bufB   = bufA + NB;
  float* bufC   = bufB + NB;
  float* bufD   = bufC + NB;
  float* deg    = bufD + NB;              // N (becomes dis in-place)
  float* wedge  = deg + N;                // E
  float* meanv  = wedge + E;              // 128
  float* rstdv  = meanv + F_IN;           // 128
  float* w3pad  = rstdv + F_IN;           // KCH*32*48 = 7680
  double* gsum  = (double*)(w3pad + (size_t)KCH * HID4 * OUTP);  // 128 doubles
  double* gsq   = gsum + F_IN;                                   // 128 doubles

  const unsigned scatterBlocks = (unsigned)(((size_t)E * 32 + THREADS - 1) / THREADS);

  // ---- preprocessing ----
  zero_f64<<<1, 256, 0, stream>>>(gsum, 2 * F_IN);
  stats_partial<<<512, F_IN, 0, stream>>>(x, gsum, gsq, N, F_IN);
  stats_final<<<1, F_IN, 0, stream>>>(gsum, gsq, meanv, rstdv, N);
  standardize<<<gblk(NB), THREADS, 0, stream>>>(x, meanv, rstdv, bufA, NB, F_IN - 1);
  zero_f32<<<gblk((size_t)N), THREADS, 0, stream>>>(deg, (size_t)N);
  degree_k<<<gblk((size_t)E), THREADS, 0, stream>>>(src, deg, E);
  dis_k<<<gblk((size_t)N), THREADS, 0, stream>>>(deg, N);
  wedge_k<<<gblk((size_t)E), THREADS, 0, stream>>>(src, dst, deg, wedge, E);
  pad_w3<<<gblk((size_t)KCH * HID4 * OUTP), THREADS, 0, stream>>>(W3, w3pad, KCH * HID4);

  // ---- one Chebyshev layer (W already padded so Fout is its leading stride) ----
  auto cheb_layer = [&](float* in, float* t1, float* t2, float* o,
                        const float* W, int Fin, int Fout) {
    const size_t nIn = (size_t)N * Fin;
    const size_t wstep = (size_t)Fin * Fout;
    const unsigned gemmBlocks =
        (unsigned)((((size_t)(N >> 4) * (Fout >> 4)) + 7) / 8);  // 8 waves/block

    // k = 0 : o = in @ W[0]
    gemm_wmma_f32<<<gemmBlocks, THREADS, 0, stream>>>(in, W, o, N, Fin, Fout, 0);
    // Tx1 = prop(in)
    zero_f32<<<gblk(nIn), THREADS, 0, stream>>>(t1, nIn);
    scatter_k<<<scatterBlocks, THREADS, 0, stream>>>(in, t1, src, dst, wedge, Fin,
                                                     1.0f, E);
    gemm_wmma_f32<<<gemmBlocks, THREADS, 0, stream>>>(t1, W + wstep, o, N, Fin,
                                                      Fout, 1);
    // k = 2..4 : Tx2 = 2*prop(Tx1) - Tx0, fused as (init Tx2=-Tx0; scatter scale=2)
    float* tx0 = in; float* tx1 = t1; float* tx2 = t2;
    for (int k = 2; k < KCH; ++k) {
      negcopy_f32<<<gblk(nIn), THREADS, 0, stream>>>(tx2, tx0, nIn);
      scatter_k<<<scatterBlocks, THREADS, 0, stream>>>(tx1, tx2, src, dst, wedge,
                                                       Fin, 2.0f, E);
      gemm_wmma_f32<<<gemmBlocks, THREADS, 0, stream>>>(tx2, W + (size_t)k * wstep,
                                                        o, N, Fin, Fout, 1);
      float* t0old = tx0; tx0 = tx1; tx1 = tx2; tx2 = t0old;
    }
  };

  // layer 1: 128 -> 128, relu
  cheb_layer(bufA, bufB, bufC, bufD, W1, F_IN, HID);
  bias_relu<<<gblk((size_t)N * HID), THREADS, 0, stream>>>(bufD, b1,
                                                           (size_t)N * HID, HID - 1);
  // layer 2: 128 -> 32, relu
  cheb_layer(bufD, bufA, bufB, bufC, W2, HID, HID4);
  bias_relu<<<gblk((size_t)N * HID4), THREADS, 0, stream>>>(bufC, b2,
                                                            (size_t)N * HID4, HID4 - 1);
  // layer 3: 32 -> 48 (pre-padded weights; cols 40..47 stay zero)
  cheb_layer(bufC, bufA, bufB, bufD, w3pad, HID4, OUTP);
  final_out<<<gblk((size_t)N * OUTF), THREADS, 0, stream>>>(bufD, b3, out, N);

  (void)n_in; (void)out_size; (void)ws_size;
}